// Graph3DBias_73564199846163
// MI455X (gfx1250) — compile-verified
//
#include <hip/hip_runtime.h>
#include <hip/hip_bf16.h>
#include <math.h>

// ---------------------------------------------------------------------------
// Graph3DBias for MI455X (gfx1250, wave32)
//   Stage 1: per-(b,i) Gaussian edge-feature sums  -> S[2048,128] (f32, in d_ws)
//   Stage 2: S @ proj_w^T + proj_b via V_WMMA_F32_16X16X4_F32 (f32 WMMA)
// ---------------------------------------------------------------------------

typedef __attribute__((ext_vector_type(2))) float v2f;
typedef __attribute__((ext_vector_type(8))) float v8f;

#define BATCH     8
#define NNODE     256
#define KBASIS    128
#define EMBED     768
#define ROWS      (BATCH * NNODE)          // 2048
#define INV_SQRT_2PI 0.39894244f           // 1/sqrt(2*3.14159)

// ---------------------------------------------------------------------------
// Kernel 1: sum_edge_features
// grid = 2048 blocks (one per (b,i)), block = 256 threads (8 waves)
// ---------------------------------------------------------------------------
__global__ __launch_bounds__(256) void edge_sum_kernel(
    const float* __restrict__ pos,     // (B, N, 3)
    const int*   __restrict__ xatom,   // (B, N)  (harness passes integers as int*)
    const float* __restrict__ means,   // (K)
    const float* __restrict__ stds,    // (K)
    const float* __restrict__ mul_w,   // (1536, 1)
    const float* __restrict__ bias_w,  // (1536, 1)
    float*       __restrict__ S)       // (2048, K) scratch out
{
    __shared__ float s_y[NNODE];       // y scalar per j
    __shared__ float s_valid[NNODE];   // 1.0 if column j not padded
    __shared__ float s_part[256];

    const int bn = blockIdx.x;         // 0..2047
    const int b  = bn >> 8;
    const int t  = threadIdx.x;        // 0..255

    // ---- phase 1: one thread per j computes the per-edge scalar ----
    const float pix = pos[bn * 3 + 0];
    const float piy = pos[bn * 3 + 1];
    const float piz = pos[bn * 3 + 2];
    const int   ai  = xatom[bn];
    {
        const int j  = t;
        const int bj = b * NNODE + j;
        const float dx = pix - pos[bj * 3 + 0];
        const float dy = piy - pos[bj * 3 + 1];
        const float dz = piz - pos[bj * 3 + 2];
        const float dist = sqrtf(dx * dx + dy * dy + dz * dz);
        const int   aj = xatom[bj];
        const int   e  = ai * 128 + aj;            // node_type_edge index
        s_y[j]     = mul_w[e] * dist + bias_w[e];
        s_valid[j] = (aj == 0) ? 0.0f : 1.0f;      // padding_mask on column j
    }
    __syncthreads();

    // ---- phase 2: 128 k-lanes x 2 j-halves accumulate Gaussians ----
    const int   k    = t & (KBASIS - 1);
    const int   jh   = t >> 7;                     // 0 or 1
    const float mean = means[k];
    const float sdev = fabsf(stds[k]) + 0.8f;
    const float inv_sdev = 1.0f / sdev;
    const float norm = inv_sdev * INV_SQRT_2PI;    // 1/(sqrt(2pi)*std)

    float acc = 0.0f;
    const int jbase = jh << 7;
#pragma unroll 8
    for (int j = jbase; j < jbase + 128; ++j) {
        const float z = (s_y[j] - mean) * inv_sdev;
        acc += s_valid[j] * __expf(-0.5f * z * z); // v_exp_f32 fast path
    }
    s_part[t] = acc * norm;
    __syncthreads();

    if (t < KBASIS) {
        S[bn * KBASIS + k] = s_part[k] + s_part[k + KBASIS];
    }
}

// ---------------------------------------------------------------------------
// Kernel 2: merge = S @ proj_w^T + proj_b  (masked), via f32 WMMA 16x16x4
// One wave per 16x16 output tile. 128 M-tiles x 48 N-tiles = 6144 waves.
// block = 256 threads (8 waves) -> grid = 768 blocks (exact cover, EXEC all-1s)
//
// Fragment layout assumptions (ISA 7.12.2, 32-bit A 16x4 table):
//   A (16x4):  lane l, m = l&15; VGPR0/1 = k = koff, koff+1 with koff = (l>=16)*2
//   B (4x16):  mirrored: lane l supplies column n = l&15, k = koff, koff+1
//   C/D:       VGPR r -> m = r + 8*(l>=16), n = l&15
// Both A and B fragments are contiguous float2 loads (k is innermost in
// S[row,k] and proj_w[d,k]).
// ---------------------------------------------------------------------------
__global__ __launch_bounds__(256) void proj_wmma_kernel(
    const float* __restrict__ S,       // (2048, 128)
    const float* __restrict__ proj_w,  // (768, 128)
    const float* __restrict__ proj_b,  // (768)
    const int*   __restrict__ xatom,   // (2048) for the row padding mask
    float*       __restrict__ out)     // (2048, 768)
{
    const int NT   = EMBED / 16;                         // 48 column tiles
    const int wave = (blockIdx.x * blockDim.x + threadIdx.x) >> 5;
    const int lane = threadIdx.x & 31;
    const int mt   = wave / NT;
    const int nt   = wave - mt * NT;
    const int row0 = mt * 16;
    const int col0 = nt * 16;

    const int hi   = lane >> 4;        // 0: lanes 0-15, 1: lanes 16-31
    const int l16  = lane & 15;
    const int koff = hi * 2;

    const float* Arow = S      + (size_t)(row0 + l16) * KBASIS + koff;
    const float* Brow = proj_w + (size_t)(col0 + l16) * KBASIS + koff;

    v8f acc = {};
#pragma unroll
    for (int k0 = 0; k0 < KBASIS; k0 += 4) {
        const v2f a = *(const v2f*)(Arow + k0);
        const v2f bv = *(const v2f*)(Brow + k0);
        // 8 args: (neg_a, A, neg_b, B, c_mod, C, reuse_a, reuse_b)
        acc = __builtin_amdgcn_wmma_f32_16x16x4_f32(
            false, a, false, bv, (short)0, acc, false, false);
    }

    const float bias = proj_b[col0 + l16];
#pragma unroll
    for (int r = 0; r < 8; ++r) {
        const int m   = r + hi * 8;
        const int row = row0 + m;
        float val = acc[r] + bias;
        if (xatom[row] == 0) val = 0.0f;   // padding_mask on output row
        out[(size_t)row * EMBED + col0 + l16] = val;
    }
}

// ---------------------------------------------------------------------------
// Host launcher
// Input order (setup_inputs): pos, x, means, stds, mul_w, bias_w, proj_w, proj_b
// ---------------------------------------------------------------------------
extern "C" void kernel_launch(void* const* d_in, const int* in_sizes, int n_in,
                              void* d_out, int out_size, void* d_ws, size_t ws_size,
                              hipStream_t stream) {
    const float* pos    = (const float*)d_in[0];
    const int*   xatom  = (const int*)  d_in[1];
    const float* means  = (const float*)d_in[2];
    const float* stds   = (const float*)d_in[3];
    const float* mul_w  = (const float*)d_in[4];
    const float* bias_w = (const float*)d_in[5];
    const float* proj_w = (const float*)d_in[6];
    const float* proj_b = (const float*)d_in[7];
    float*       out    = (float*)d_out;
    float*       S      = (float*)d_ws;            // 2048*128*4 = 1 MB scratch

    edge_sum_kernel<<<ROWS, 256, 0, stream>>>(pos, xatom, means, stds,
                                              mul_w, bias_w, S);

    const int tiles  = (ROWS / 16) * (EMBED / 16); // 6144 wave-tiles
    const int blocks = tiles / 8;                  // 8 waves per 256-thr block
    proj_wmma_kernel<<<blocks, 256, 0, stream>>>(S, proj_w, proj_b, xatom, out);
}